// KMeans_23862838297106
// MI455X (gfx1250) — compile-verified
//
#include <hip/hip_runtime.h>

// KMeans predict, MI455X (gfx1250, wave32).
// Compute-bound fp32 problem (AI ~500 FLOP/B) -> V_WMMA_F32_16X16X4_F32.
// Fused GEMM + argmin: the NxK score matrix never hits memory.
// Centers double-buffered in LDS via GLOBAL_LOAD_ASYNC_TO_LDS (ASYNCcnt path).

typedef __attribute__((ext_vector_type(2))) float v2f;
typedef __attribute__((ext_vector_type(8))) float v8f;
typedef __attribute__((ext_vector_type(4))) int   v4i;   // b128 payload type

#define DIMS        64      // D
#define KSTEPS      16      // D / 4 (WMMA k=4 per step)
#define CHUNK       64      // centers staged per LDS chunk
#define LDS_STRIDE  68      // padded floats per center row (68*4=272=17*16 bytes:
                            //  16B alignment for b128 stores, and b64 B-fragment
                            //  reads spread across all 64 LDS banks)
#define BLOCK_ROWS  128     // 8 waves * 16 rows
#define NTHREADS    256

#if defined(__has_builtin)
# if __has_builtin(__builtin_amdgcn_global_load_async_to_lds_b128) && \
     __has_builtin(__builtin_amdgcn_s_wait_asynccnt)
#  define HAVE_ASYNC_LDS 1
# endif
#endif
#ifndef HAVE_ASYNC_LDS
# define HAVE_ASYNC_LDS 0
#endif

#define LDS_AS  __attribute__((address_space(3)))
#define GLOB_AS __attribute__((address_space(1)))

// Generic->LDS: generic LDS pointers carry the LDS byte offset in the low 32
// bits.
static __device__ __forceinline__ LDS_AS v4i* as_lds_b128(void* p) {
    return (LDS_AS v4i*)(unsigned)(unsigned long long)p;
}
// Builtin wants the global source typed as a b128 vector pointer in AS(1).
static __device__ __forceinline__ GLOB_AS v4i* as_glob_b128(const void* p) {
    return (GLOB_AS v4i*)(unsigned long long)p;
}

// ---- prologue: m2[k] = ||centers[k]||^2 ----
__global__ void kmeans_m2_kernel(const float* __restrict__ centers,
                                 float* __restrict__ m2, int K) {
    int k = blockIdx.x * blockDim.x + threadIdx.x;
    if (k < K) {
        const float* c = centers + (size_t)k * DIMS;
        float s = 0.0f;
#pragma unroll
        for (int i = 0; i < DIMS; ++i) s = fmaf(c[i], c[i], s);
        m2[k] = s;
    }
}

__global__ __launch_bounds__(NTHREADS)
void kmeans_wmma_kernel(const float* __restrict__ x,
                        const float* __restrict__ centers,
                        const float* __restrict__ m2,
                        int* __restrict__ out,
                        int n, int K) {
    __shared__ __align__(16) float cbuf[2][CHUNK * LDS_STRIDE];
    __shared__ __align__(16) float m2buf[2][CHUNK];

    const int lane = threadIdx.x & 31;
    const int wave = threadIdx.x >> 5;
    const int half = lane >> 4;      // 0: K/M group lo, 1: hi
    const int ln   = lane & 15;
    const int koff = half * 2;

    const int rbase = blockIdx.x * BLOCK_ROWS + wave * 16;

    // ---- A fragments: 16 rows of x, all of D, kept in registers ----
    // ISA 16x4 fp32 A layout: lane(0-15)=M, vgpr pair = {K, K+1}, lanes 16-31 = K+2.
    int arow = rbase + ln;
    if (arow >= n) arow = n - 1;                 // clamp; stores are guarded
    const float* xrow = x + (size_t)arow * DIMS;
    v2f A[KSTEPS];
#pragma unroll
    for (int kt = 0; kt < KSTEPS; ++kt)
        A[kt] = *(const v2f*)(xrow + kt * 4 + koff);

    // running argmin per (row-vgpr, col-lane)
    float best[8];
    int   bidx[8];
#pragma unroll
    for (int v = 0; v < 8; ++v) { best[v] = 3.4e38f; bidx[v] = 0; }

    // cooperative stage of one 64-center chunk (+ its m2 slice) into padded LDS
    auto stage = [&](int c, int b) {
        const float* src = centers + (size_t)c * CHUNK * DIMS;
#if HAVE_ASYNC_LDS
#pragma unroll
        for (int i = 0; i < (CHUNK * DIMS / 4) / NTHREADS; ++i) {
            int f   = threadIdx.x + i * NTHREADS;  // float4 index in chunk
            int row = f >> 4;                      // 16 float4 per center row
            int c4  = f & 15;
            __builtin_amdgcn_global_load_async_to_lds_b128(
                as_glob_b128(src + f * 4),
                as_lds_b128(&cbuf[b][row * LDS_STRIDE + c4 * 4]),
                /*offset=*/0, /*cpol=*/0);
        }
        if (threadIdx.x < CHUNK / 4) {
            __builtin_amdgcn_global_load_async_to_lds_b128(
                as_glob_b128(m2 + (size_t)c * CHUNK + threadIdx.x * 4),
                as_lds_b128(&m2buf[b][threadIdx.x * 4]),
                /*offset=*/0, /*cpol=*/0);
        }
#else
#pragma unroll
        for (int i = 0; i < (CHUNK * DIMS / 4) / NTHREADS; ++i) {
            int f   = threadIdx.x + i * NTHREADS;
            int row = f >> 4;
            int c4  = f & 15;
            float4 val = ((const float4*)src)[f];
            *(float4*)&cbuf[b][row * LDS_STRIDE + c4 * 4] = val;
        }
        if (threadIdx.x < CHUNK / 4) {
            float4 val = ((const float4*)(m2 + (size_t)c * CHUNK))[threadIdx.x];
            *(float4*)&m2buf[b][threadIdx.x * 4] = val;
        }
#endif
    };

    stage(0, 0);

    const int nchunks = K / CHUNK;
    for (int c = 0; c < nchunks; ++c) {
#if HAVE_ASYNC_LDS
        __builtin_amdgcn_s_wait_asynccnt(0);       // my chunk-c async writes landed
#endif
        __syncthreads();                           // chunk c visible to all waves
        const float* cb = cbuf[c & 1];
        if (c + 1 < nchunks) stage(c + 1, (c + 1) & 1);  // overlap next stage

        // pre-read this chunk's m2 values so no load sits in the WMMA chain
        float m2t[CHUNK / 16];
#pragma unroll
        for (int t = 0; t < CHUNK / 16; ++t)
            m2t[t] = m2buf[c & 1][t * 16 + ln];

#pragma unroll
        for (int t = 0; t < CHUNK / 16; ++t) {
            // B fragment base: lane(0-15)=N (center in tile), vgpr pair = {K, K+1}
            const float* bp = cb + (size_t)(t * 16 + ln) * LDS_STRIDE + koff;
            v8f acc = {0.f, 0.f, 0.f, 0.f, 0.f, 0.f, 0.f, 0.f};
#pragma unroll
            for (int kt = 0; kt < KSTEPS; ++kt) {
                v2f B = *(const v2f*)(bp + kt * 4);
                acc = __builtin_amdgcn_wmma_f32_16x16x4_f32(
                    /*neg_a=*/false, A[kt], /*neg_b=*/false, B,
                    /*c_mod=*/(short)0, acc, /*reuse_a=*/false, /*reuse_b=*/false);
            }
            const int cidx = c * CHUNK + t * 16 + ln;
#pragma unroll
            for (int v = 0; v < 8; ++v) {
                float s = fmaf(-2.0f, acc[v], m2t[t]);
                if (s < best[v]) { best[v] = s; bidx[v] = cidx; }
            }
        }
    }

    // ---- per-row argmin across the 16 column lanes (stays within 16-lane halves) ----
#pragma unroll
    for (int v = 0; v < 8; ++v) {
        float bv = best[v];
        int   bi = bidx[v];
#pragma unroll
        for (int m = 8; m >= 1; m >>= 1) {
            float ov = __shfl_xor(bv, m, 32);
            int   oi = __shfl_xor(bi, m, 32);
            if (ov < bv || (ov == bv && oi < bi)) { bv = ov; bi = oi; }
        }
        // lane 0 holds rows v (M=v), lane 16 holds rows v+8 (M=v+8)
        int row = rbase + half * 8 + v;
        if (ln == 0 && row < n) out[row] = bi;
    }
}

extern "C" void kernel_launch(void* const* d_in, const int* in_sizes, int n_in,
                              void* d_out, int out_size, void* d_ws, size_t ws_size,
                              hipStream_t stream) {
    const float* x       = (const float*)d_in[0];
    const float* centers = (const float*)d_in[1];
    const int n = in_sizes[0] / DIMS;
    const int K = in_sizes[1] / DIMS;

    float* m2 = (float*)d_ws;          // K floats of scratch
    int*  out = (int*)d_out;           // argmin ids (int32)

    kmeans_m2_kernel<<<(K + 255) / 256, 256, 0, stream>>>(centers, m2, K);

    const int blocks = (n + BLOCK_ROWS - 1) / BLOCK_ROWS;
    kmeans_wmma_kernel<<<blocks, NTHREADS, 0, stream>>>(x, centers, m2, out, n, K);
}